// FFA_54528904790771
// MI455X (gfx1250) — compile-verified
//
#include <hip/hip_runtime.h>
#include <math.h>

// ---------------------------------------------------------------------------
// FFA forward for MI455X (gfx1250, wave32).
//   out[b,t,m,c] = sum_{s<=t} lam^{t-s} x[b,s,m] + lam^{t+1} memory[b,m,c]
//   lam_{m,c} = exp(clip(a_m)) * (cos b_c + i sin b_c)
// 3-level scan: 16-wide blocks via V_WMMA_F32_16X16X4_F32, 128-block serial
// scan, fused carry correction + LDS-transposed coalesced store.
// x tile is staged to LDS once per workgroup with GLOBAL_LOAD_ASYNC_TO_LDS
// (ASYNCcnt / s_wait_asynccnt path).
// ---------------------------------------------------------------------------

#define BB    8
#define TT    2048
#define MEMD  64
#define CTXD  16
#define BLKW  16
#define NBLK  128          // TT / BLKW
#define LIMITF 0.07664339751178551f   // log(FLT_MAX)/1024 - 0.01

typedef __attribute__((ext_vector_type(2))) float v2f;
typedef __attribute__((ext_vector_type(8))) float v8f;

__device__ __forceinline__ float clip_a(float av) {
    return fminf(fmaxf(av, -LIMITF), -1e-8f);
}

// -------------------------------------------------------------------------
// Kernel 1: block sums  S[(b,m,c),blk] = sum_{s=0..15} lam^{15-s} x[b,blk*16+s,m]
// Fully parallel (no long chain): 1,048,576 threads, 16-step Horner each.
// -------------------------------------------------------------------------
__global__ void ffa_blocksum(const float* __restrict__ x,
                             const float* __restrict__ a,
                             const float* __restrict__ bv_,
                             float2* __restrict__ S) {
    int tid  = blockIdx.x * blockDim.x + threadIdx.x;   // 0 .. 1048575
    int blk  = tid & (NBLK - 1);
    int rest = tid >> 7;                                // (b*64+m)*16+c
    int c    = rest & (CTXD - 1);
    int m    = (rest >> 4) & (MEMD - 1);
    int b    = rest >> 10;

    float av = clip_a(a[m]);
    float bv = bv_[c];
    float e  = __expf(av);
    float lr = e * __cosf(bv), li = e * __sinf(bv);

    const float* xp = x + (size_t)((b * TT + blk * BLKW) * MEMD + m);
    float hr = 0.f, hi = 0.f;
#pragma unroll
    for (int s = 0; s < BLKW; ++s) {
        float xs = xp[s * MEMD];
        float nr = fmaf(lr, hr, fmaf(-li, hi, xs));
        float ni = fmaf(lr, hi, li * hr);
        hr = nr; hi = ni;
    }
    S[tid] = make_float2(hr, hi);
}

// -------------------------------------------------------------------------
// Kernel 2: inter-block scan (chain length 128, 8192 independent threads)
//   h_in[blk] = state entering block blk;  h <- lam^16 * h + S[blk]
// -------------------------------------------------------------------------
__global__ void ffa_scan(const float2* __restrict__ S,
                         const float* __restrict__ mr,
                         const float* __restrict__ mi,
                         const float* __restrict__ a,
                         const float* __restrict__ bv_,
                         float2* __restrict__ hin) {
    int tid = blockIdx.x * blockDim.x + threadIdx.x;    // 0 .. 8191 = (b*64+m)*16+c
    int c = tid & (CTXD - 1);
    int m = (tid >> 4) & (MEMD - 1);

    float av = clip_a(a[m]);
    float bv = bv_[c];
    float e16 = __expf(16.f * av);
    float lr = e16 * __cosf(16.f * bv), li = e16 * __sinf(16.f * bv);

    float hr = mr[tid], hi = mi[tid];
    const float2* Sp = S + (size_t)tid * NBLK;
    float2*       Hp = hin + (size_t)tid * NBLK;
    for (int k = 0; k < NBLK; ++k) {
        Hp[k] = make_float2(hr, hi);
        float2 s = Sp[k];
        float nr = fmaf(lr, hr, fmaf(-li, hi, s.x));
        float ni = fmaf(lr, hi, fmaf(li, hr, s.y));
        hr = nr; hi = ni;
    }
}

// -------------------------------------------------------------------------
// Kernel 3: WMMA intra-block prefix + carry correction + coalesced store.
// Workgroup = one m, one b (== blockIdx.y), 16 waves (one per c).
// Its 8 tiles cover blk 0..127, so the x footprint is exactly x[b,:,m] (8KB)
// -> staged into LDS once via GLOBAL_LOAD_ASYNC_TO_LDS_B32, then B operands
// come from aligned ds_load_b64.
// -------------------------------------------------------------------------
__global__ __launch_bounds__(512) void ffa_wmma(const float* __restrict__ x,
                                                const float* __restrict__ a,
                                                const float* __restrict__ bv_,
                                                const float2* __restrict__ hin,
                                                float2* __restrict__ out) {
    __shared__ float  xs_lds[TT];                       // x[b, :, m]   (8 KB)
    __shared__ float2 tileLds[BLKW * 16 * CTXD];        // [t][col][c] (32 KB)

    int m    = blockIdx.x;           // 0..63
    int b    = blockIdx.y;           // 0..7  (batch == column-tile group)
    int tidx = threadIdx.x;          // 0..511
    int lane = tidx & 31;
    int c    = tidx >> 5;            // wave id == ctx index
    int M    = lane & 15;
    bool lo  = lane < 16;
    int  off = lo ? 0 : 8;

    // ---- async-stage x[b, :, m] into LDS (CDNA5 async-to-LDS path) ----
#pragma unroll
    for (int i = tidx; i < TT; i += 512) {
        const float* gp = x + (size_t)((b * TT + i) * MEMD + m);
        unsigned int  lds_addr = (unsigned int)(unsigned long long)(&xs_lds[i]);
        asm volatile("global_load_async_to_lds_b32 %0, %1, off"
                     :: "v"(lds_addr), "v"(gp)
                     : "memory");
    }

    float av = clip_a(a[m]);
    float bv = bv_[c];

    // A operands: L_r / L_i, 16x16 lower-triangular, ISA f32 A layout:
    //   VGPR0 = {K=4k+0 | K=4k+2}, VGPR1 = {K=4k+1 | K=4k+3}, row M = lane%16
    v2f Ar[4], Ai[4];
#pragma unroll
    for (int k = 0; k < 4; ++k) {
        int kk = 4 * k + (lo ? 0 : 2);
        float e0r = 0.f, e0i = 0.f, e1r = 0.f, e1i = 0.f;
        int d0 = M - kk;
        if (d0 >= 0) { float e = __expf(av * (float)d0);
                       e0r = e * __cosf(bv * (float)d0);
                       e0i = e * __sinf(bv * (float)d0); }
        int d1 = d0 - 1;
        if (d1 >= 0) { float e = __expf(av * (float)d1);
                       e1r = e * __cosf(bv * (float)d1);
                       e1i = e * __sinf(bv * (float)d1); }
        Ar[k] = (v2f){e0r, e1r};
        Ai[k] = (v2f){e0i, e1i};
    }

    // Carry weights w[r] = lam^{t+1}, t = r + off  (depends only on lane)
    float wr[8], wi[8];
#pragma unroll
    for (int r = 0; r < 8; ++r) {
        float t1 = (float)(r + off + 1);
        float e = __expf(av * t1);
        wr[r] = e * __cosf(bv * t1);
        wi[r] = e * __sinf(bv * t1);
    }

    // drain this wave's async loads, then make LDS visible to all waves
    asm volatile("s_wait_asynccnt 0x0" ::: "memory");
    __syncthreads();

    for (int tile = 0; tile < 8; ++tile) {
        int blkBase = tile << 4;             // first block of this 16-col tile
        int blk     = blkBase + M;           // this lane's column's block

        v8f Dr = {0.f,0.f,0.f,0.f,0.f,0.f,0.f,0.f};
        v8f Di = {0.f,0.f,0.f,0.f,0.f,0.f,0.f,0.f};
#pragma unroll
        for (int k = 0; k < 4; ++k) {
            int s0 = 4 * k + (lo ? 0 : 2);   // B f32 layout mirrors A: K0/K2 in .x
            float2 bx = *(const float2*)&xs_lds[blk * BLKW + s0];  // ds_load_b64
            v2f Bx = {bx.x, bx.y};
            Dr = __builtin_amdgcn_wmma_f32_16x16x4_f32(false, Ar[k], false, Bx,
                                                       (short)0, Dr, false, false);
            Di = __builtin_amdgcn_wmma_f32_16x16x4_f32(false, Ai[k], false, Bx,
                                                       (short)0, Di, false, false);
        }

        float2 h = hin[(size_t)(((b * MEMD + m) * CTXD + c) * NBLK + blk)];

        __syncthreads();   // previous tile's LDS reads done
#pragma unroll
        for (int r = 0; r < 8; ++r) {
            int t = r + off;                 // D layout: VGPR r -> M = r (+8 hi lanes)
            float orr = Dr[r] + wr[r] * h.x - wi[r] * h.y;
            float oii = Di[r] + wr[r] * h.y + wi[r] * h.x;
            tileLds[(t * 16 + M) * CTXD + c] = make_float2(orr, oii);
        }
        __syncthreads();

        // Cooperative coalesced store: 4096 complex, 512 threads, c fastest ->
        // 16 lanes x 8B = 128B contiguous per (b,t,m).
#pragma unroll
        for (int it = 0; it < 8; ++it) {
            int p    = it * 512 + tidx;
            int c2   = p & 15;
            int col2 = (p >> 4) & 15;
            int t2   = p >> 8;
            size_t gidx = (size_t)(((b * TT + (blkBase + col2) * BLKW + t2) * MEMD + m)
                                   * CTXD + c2);
            out[gidx] = tileLds[(t2 * 16 + col2) * CTXD + c2];
        }
    }
}

// ---------------------------------------------------------------------------
extern "C" void kernel_launch(void* const* d_in, const int* in_sizes, int n_in,
                              void* d_out, int out_size, void* d_ws, size_t ws_size,
                              hipStream_t stream) {
    const float* x  = (const float*)d_in[0];   // (8,2048,64)
    const float* mr = (const float*)d_in[1];   // (8,1,64,16)
    const float* mi = (const float*)d_in[2];   // (8,1,64,16)
    const float* a  = (const float*)d_in[3];   // (1,1,64)
    const float* bb = (const float*)d_in[4];   // (1,1,16)
    float2* out = (float2*)d_out;              // complex64 interleaved

    const int NCHAIN = BB * MEMD * CTXD;                 // 8192
    float2* S   = (float2*)d_ws;                         // 8 MiB
    float2* hin = S + (size_t)NCHAIN * NBLK;             // 8 MiB

    ffa_blocksum<<<(NCHAIN * NBLK) / 256, 256, 0, stream>>>(x, a, bb, S);
    ffa_scan<<<NCHAIN / 256, 256, 0, stream>>>(S, mr, mi, a, bb, hin);
    ffa_wmma<<<dim3(MEMD, BB), 512, 0, stream>>>(x, a, bb, hin, out);
}